// KMeans_69595650064393
// MI455X (gfx1250) — compile-verified
//
#include <hip/hip_runtime.h>
#include <hip/hip_bf16.h>

// KMeans distances + argmin via V_WMMA_F32_16X16X4_F32 on gfx1250 (wave32).
//
// distances[k][n] = ||c_k||^2 + ||x_n||^2 - 2 c_k . x_n
// The K-dim of the GEMM is augmented so the WMMA accumulator IS the distance:
//   A'[k] = [ c_k(0..127), ||c_k||^2, 1, 0, 0 ]          (132 dims)
//   B'[n] = [ -2*x_n(0..127), 1, ||x_n||^2, 0, 0 ]       (132 dims)
//   A'.B' = -2 c.x + ||c||^2 + ||x||^2 = distance
// argmin over k is fused in-register (8 acc VGPRs per lane -> xor-16 shuffle),
// avoiding a second 0.5 GB pass over the distance matrix.
//
// Each wave owns TWO 16-column N-tiles: the A (centroid) fragment is loaded
// once and fed to two WMMAs (3 ds_load_b64 per 2 WMMAs), and the two
// independent accumulator chains give the XDL pipe interleavable work.

typedef __attribute__((ext_vector_type(2))) float v2f;
typedef __attribute__((ext_vector_type(8))) float v8f;

static constexpr int N = 262144;
static constexpr int D = 128;
static constexpr int K = 512;
static constexpr int LDSTR = 132;   // 128 data + [csq|1][1|xsq][0][0]; 132%64=4 -> conflict-free

__global__ __launch_bounds__(256) void kmeans_wmma_kernel(
    const float* __restrict__ inputs,    // (N, D) row-major
    const float* __restrict__ cents,     // (K, D) row-major
    float* __restrict__ out)             // [K*N distances] ++ [N assignments-as-float]
{
    __shared__ float ldsX[256 * LDSTR];  // -2x tiles, augmented (135 KB)
    __shared__ float ldsC[16  * LDSTR];  // centroid tile, augmented (8.25 KB)

    const int t     = threadIdx.x;
    const int lane  = t & 31;
    const int wave  = t >> 5;            // 8 waves; wave owns 32 N-columns (2 tiles)
    const int nbase = blockIdx.x * 256;

    // ---- Stage input tile: 256 columns x 128 dims. One full row per thread:
    //      store -2x, fold ||x||^2 in-register, write augmentation slots.
    {
        const float4* g = (const float4*)(inputs + (long long)(nbase + t) * D);
        float4* dst = (float4*)&ldsX[t * LDSTR];
        float xsq = 0.0f;
        #pragma unroll
        for (int i = 0; i < 32; ++i) {
            float4 v = g[i];
            xsq += v.x * v.x + v.y * v.y + v.z * v.z + v.w * v.w;
            dst[i] = make_float4(-2.0f * v.x, -2.0f * v.y, -2.0f * v.z, -2.0f * v.w);
        }
        dst[32] = make_float4(1.0f, xsq, 0.0f, 0.0f);  // pairs with A's [csq, 1, 0, 0]
    }
    __syncthreads();

    // Per-lane fragment addressing (wave32, 16x16x4 f32):
    //   A (16x4): lanes 0-15 -> M=lane, K=d0+{0,1}; lanes 16-31 -> M=lane-16, K=d0+{2,3}
    //   B (4x16): mirrored -> lanes 0-15 hold rows K=d0+{0,1}, lanes 16-31 rows d0+{2,3}
    const int m    = lane & 15;
    const int ksel = (lane >> 4) * 2;
    const float* aRow  = &ldsC[m * LDSTR + ksel];
    const float* bRow0 = &ldsX[(wave * 32 + m)      * LDSTR + ksel];
    const float* bRow1 = &ldsX[(wave * 32 + 16 + m) * LDSTR + ksel];
    const int n0 = nbase + wave * 32 + m;    // this lane's two output columns
    const int n1 = n0 + 16;

    float bestv0 = 3.402823466e+38f, bestv1 = 3.402823466e+38f;
    int   besti0 = 0,                besti1 = 0;

    for (int kt = 0; kt < K / 16; ++kt) {
        __syncthreads();  // previous tile's WMMA reads done before overwrite

        // ---- Stage centroid tile: 16 rows x 128 dims (raw c).
        {
            const int row = t >> 4;      // 0..15
            const int seg = t & 15;      // 8 floats each
            const float4* g = (const float4*)(cents + (long long)(kt * 16 + row) * D + seg * 8);
            float4* dst = (float4*)&ldsC[row * LDSTR + seg * 8];
            dst[0] = g[0];
            dst[1] = g[1];
        }
        __syncthreads();
        if (t < 16) {
            float cs = 0.0f;
            for (int d = 0; d < D; ++d) { float c = ldsC[t * LDSTR + d]; cs += c * c; }
            ldsC[t * LDSTR + 128] = cs;     // pairs with B's 1
            ldsC[t * LDSTR + 129] = 1.0f;   // pairs with B's ||x||^2
            ldsC[t * LDSTR + 130] = 0.0f;
            ldsC[t * LDSTR + 131] = 0.0f;
        }
        __syncthreads();

        // ---- 2x33 WMMA steps over the augmented 132-dim K axis.
        //      A loaded once per step, shared by both N-tiles.
        v8f acc0 = {};
        v8f acc1 = {};
        #pragma unroll
        for (int d0 = 0; d0 < LDSTR; d0 += 4) {
            v2f a  = *(const v2f*)(aRow  + d0);  // ds_load_b64, conflict-free
            v2f b0 = *(const v2f*)(bRow0 + d0);
            v2f b1 = *(const v2f*)(bRow1 + d0);
            acc0 = __builtin_amdgcn_wmma_f32_16x16x4_f32(
                false, a, false, b0, (short)0, acc0, false, false);
            acc1 = __builtin_amdgcn_wmma_f32_16x16x4_f32(
                false, a, false, b1, (short)0, acc1, false, false);
        }

        // ---- Fused argmin + distance-tile stores (M = r + 8*(lane>=16)).
        const int mbase = kt * 16 + 8 * (lane >> 4);

        // Tile 0
        {
            float mv = acc0[0];
            int   mi = mbase;
            #pragma unroll
            for (int r = 1; r < 8; ++r) {
                float v = acc0[r];
                if (v < mv) { mv = v; mi = mbase + r; }   // ascending k: keeps first min
            }
            float ov = __shfl_xor(mv, 16, 32);
            int   oi = __shfl_xor(mi, 16, 32);
            if (ov < mv || (ov == mv && oi < mi)) { mv = ov; mi = oi; }
            if (mv < bestv0 || (mv == bestv0 && mi < besti0)) { bestv0 = mv; besti0 = mi; }
            #pragma unroll
            for (int r = 0; r < 8; ++r)
                out[(long long)(mbase + r) * N + n0] = acc0[r];
        }
        // Tile 1
        {
            float mv = acc1[0];
            int   mi = mbase;
            #pragma unroll
            for (int r = 1; r < 8; ++r) {
                float v = acc1[r];
                if (v < mv) { mv = v; mi = mbase + r; }
            }
            float ov = __shfl_xor(mv, 16, 32);
            int   oi = __shfl_xor(mi, 16, 32);
            if (ov < mv || (ov == mv && oi < mi)) { mv = ov; mi = oi; }
            if (mv < bestv1 || (mv == bestv1 && mi < besti1)) { bestv1 = mv; besti1 = mi; }
            #pragma unroll
            for (int r = 0; r < 8; ++r)
                out[(long long)(mbase + r) * N + n1] = acc1[r];
        }
    }

    // ---- Assignments (stored as float, after the K*N distance block).
    if (lane < 16) {
        out[(long long)K * N + n0] = (float)besti0;
        out[(long long)K * N + n1] = (float)besti1;
    }
}

extern "C" void kernel_launch(void* const* d_in, const int* in_sizes, int n_in,
                              void* d_out, int out_size, void* d_ws, size_t ws_size,
                              hipStream_t stream) {
    const float* inputs = (const float*)d_in[0];   // (N, D) f32
    const float* cents  = (const float*)d_in[1];   // (K, D) f32
    float* out = (float*)d_out;                    // K*N + N floats
    (void)in_sizes; (void)n_in; (void)out_size; (void)d_ws; (void)ws_size;

    dim3 grid(N / 256);   // 1024 blocks x 256 threads (8 wave32 / block)
    kmeans_wmma_kernel<<<grid, dim3(256), 0, stream>>>(inputs, cents, out);
}